// gmm_layer_87995289960657
// MI455X (gfx1250) — compile-verified
//
#include <hip/hip_runtime.h>

#define GMM_B 32768
#define GMM_K 64
#define GMM_D 64
#define GMM_EPS 1e-4f

typedef float v2f __attribute__((ext_vector_type(2)));
typedef float v8f __attribute__((ext_vector_type(8)));

// ---------------------------------------------------------------------------
// Setup: per class k, build M_k = diag(D^-1/2) * L_inv, v_k = M_k c_k,
// det_scale_k, and pre-swizzle M_k into WMMA A-fragment order.
// A fragment layout for V_WMMA_F32_16X16X4_F32 (A is 16x4 per step):
//   lane l, vgpr v  ->  A[row = l%16, col = 2*(l/16) + v]
// Fragment q (float2): lane = q%32, s = (q/32)%16 (d-chunk), t = q/512 (e-tile)
//   element = M[t*16 + lane%16][4*s + 2*(lane/16) + {0,1}]
// ---------------------------------------------------------------------------
__global__ __launch_bounds__(64) void gmm_setup(const float* __restrict__ centers,
                                                const float* __restrict__ Lmat,
                                                float* __restrict__ Aw,
                                                float* __restrict__ vvec,
                                                float* __restrict__ dscale)
{
    __shared__ float Ls[64 * 65];
    __shared__ float Li[64 * 65];
    __shared__ float Dsh[64];
    __shared__ float rssh[64];
    __shared__ float csh[64];

    const int k   = blockIdx.x;
    const int tid = threadIdx.x;   // 0..63

    // Stage unit-lower-triangular L, D = diag(Lmat)^2 + eps, centers row.
    const float* lrow = Lmat + (size_t)k * GMM_D * GMM_D + (size_t)tid * GMM_D;
    for (int j = 0; j < 64; ++j) {
        float v = lrow[j];
        Ls[tid * 65 + j] = (j < tid) ? v : ((j == tid) ? 1.0f : 0.0f);
    }
    float rd = lrow[tid];
    float Dv = rd * rd + GMM_EPS;
    Dsh[tid]  = Dv;
    rssh[tid] = rsqrtf(Dv);
    csh[tid]  = centers[k * GMM_D + tid];
    __syncthreads();

    // Forward substitution: thread j computes column j of L_inv (depends only
    // on its own column -> no intra-solve sync needed).
    {
        const int j = tid;
        for (int i = 0; i < j; ++i) Li[i * 65 + j] = 0.0f;
        Li[j * 65 + j] = 1.0f;
        for (int i = j + 1; i < 64; ++i) {
            float acc = 0.0f;
            for (int m = j; m < i; ++m) acc += Ls[i * 65 + m] * Li[m * 65 + j];
            Li[i * 65 + j] = -acc;
        }
    }
    __syncthreads();

    // v_k[e] = rs[e] * sum_d L_inv[e][d] * c[d]
    {
        float acc = 0.0f;
        for (int d = 0; d < 64; ++d) acc += Li[tid * 65 + d] * csh[d];
        vvec[k * GMM_D + tid] = acc * rssh[tid];
    }
    // det_scale = rsqrt(prod(D))  (f32 prod like the reference)
    if (tid == 0) {
        float p = 1.0f;
        for (int i = 0; i < 64; ++i) p *= Dsh[i];
        dscale[k] = rsqrtf(p);
    }

    // Pre-swizzled WMMA A fragments of M = diag(rs) * L_inv.
    v2f* awk = ((v2f*)Aw) + (size_t)k * 2048;
    for (int q = tid; q < 2048; q += 64) {
        int lane = q & 31;
        int s    = (q >> 5) & 15;
        int t    = q >> 9;
        int e    = t * 16 + (lane & 15);
        int d    = 4 * s + 2 * (lane >> 4);
        float r  = rssh[e];
        v2f val;
        val.x = Li[e * 65 + d]     * r;
        val.y = Li[e * 65 + d + 1] * r;
        awk[q] = val;
    }
}

// ---------------------------------------------------------------------------
// Main: one wave32 per 16-row batch tile. Computes U^T[e][b] = M_k X^T via
// f32 WMMA (D-tile rows = e, cols = b), reduces (u - v_k)^2 over e, then a
// per-row det-scaled softmax over the 64 classes.
// ---------------------------------------------------------------------------
__global__ __launch_bounds__(256) void gmm_main(const float* __restrict__ x,
                                                const float* __restrict__ Aw,
                                                const float* __restrict__ vvec,
                                                const float* __restrict__ dscale,
                                                float* __restrict__ out)
{
    __shared__ float elds[8 * 16 * 65];   // [wave][b row][k] padded

    const int lane = threadIdx.x & 31;
    const int wave = threadIdx.x >> 5;
    const int half = lane >> 4;           // 0 or 1
    const int lrow = lane & 15;
    const int b0   = (blockIdx.x * 8 + wave) * 16;

    // Preload B fragments (X^T tile, 64 x 16), reused for all 64 classes.
    // B 4x16 step layout: lane l, vgpr v -> B[row = 2*(l/16)+v, col = l%16]
    v2f bfrag[16];
    const float* xrow = x + (size_t)(b0 + lrow) * GMM_D + 2 * half;
#pragma unroll
    for (int s = 0; s < 16; ++s)
        bfrag[s] = *(const v2f*)(xrow + 4 * s);

    const float* abase = Aw + lane * 2;
    const int    wbase = wave * (16 * 65);

#pragma unroll 1
    for (int k = 0; k < GMM_K; ++k) {
        float part = 0.0f;
#pragma unroll
        for (int t = 0; t < 4; ++t) {
            v8f acc0 = {};
            v8f acc1 = {};
            const float* ap = abase + (size_t)(k * 4 + t) * 1024;
#pragma unroll
            for (int s = 0; s < 16; s += 2) {
                v2f a0 = *(const v2f*)(ap + s * 64);
                v2f a1 = *(const v2f*)(ap + s * 64 + 64);
                acc0 = __builtin_amdgcn_wmma_f32_16x16x4_f32(
                    false, a0, false, bfrag[s],     (short)0, acc0, false, false);
                acc1 = __builtin_amdgcn_wmma_f32_16x16x4_f32(
                    false, a1, false, bfrag[s + 1], (short)0, acc1, false, false);
            }
            const float* vp = vvec + k * GMM_D + t * 16 + 8 * half;
#pragma unroll
            for (int r = 0; r < 8; ++r) {
                float u  = acc0[r] + acc1[r];
                float dd = u - vp[r];
                part = fmaf(dd, dd, part);
            }
        }
        // Combine the two half-wave partials (lanes l and l^16): SWAPX16.
        int sw = __builtin_amdgcn_ds_swizzle(__float_as_int(part), 0x401f);
        part += __int_as_float(sw);
        if (half == 0) elds[wbase + lrow * 65 + k] = -0.5f * part;
    }
    __syncthreads();

    // Softmax over classes for batch row (b0 + lrow); both halves compute it,
    // each half stores its 32 output columns.
    const float* myrow = elds + wbase + lrow * 65;
    float mx = myrow[0];
    for (int k = 1; k < GMM_K; ++k) mx = fmaxf(mx, myrow[k]);
    float denom = 0.0f;
    for (int k = 0; k < GMM_K; ++k)
        denom += dscale[k] * __expf(myrow[k] - mx);
    float invd = 1.0f / denom;

    float* orow = out + (size_t)(b0 + lrow) * GMM_K + half * 32;
    for (int kk = 0; kk < 32; ++kk) {
        int k = half * 32 + kk;
        orow[kk] = dscale[k] * __expf(myrow[k] - mx) * invd;
    }
}

// ---------------------------------------------------------------------------
extern "C" void kernel_launch(void* const* d_in, const int* in_sizes, int n_in,
                              void* d_out, int out_size, void* d_ws, size_t ws_size,
                              hipStream_t stream)
{
    const float* x       = (const float*)d_in[0];   // (32768, 64)
    const float* centers = (const float*)d_in[1];   // (64, 64)
    const float* Lmat    = (const float*)d_in[2];   // (64, 64, 64)
    float*       out     = (float*)d_out;           // (32768, 64)

    // Workspace: Aw (64*4096 f32 = 1 MB) | vvec (4096 f32) | dscale (64 f32)
    float* Aw = (float*)d_ws;
    float* vv = Aw + 64 * 4096;
    float* ds = vv + 64 * 64;

    gmm_setup<<<GMM_K, 64, 0, stream>>>(centers, Lmat, Aw, vv, ds);
    gmm_main<<<GMM_B / 128, 256, 0, stream>>>(x, Aw, vv, ds, out);
}